// LUT_231928234067
// MI455X (gfx1250) — compile-verified
//
#include <hip/hip_runtime.h>

typedef __attribute__((ext_vector_type(2))) float v2f;
typedef __attribute__((ext_vector_type(8))) float v8f;

#define HW_IMG (1024*1024)
#define NPIX   1024      // 32*32 spatial positions per batch
#define CIN    1024
#define COUT   256
#define LUT3   35937     // 33^3
#define LUTC   (3*LUT3)  // 107811 floats per (batch) blended LUT

// GEMM tiling
#define BLK_M  64
#define BLK_N  128
#define BLK_K  32
#define W_PAD  36        // 64-row W slab row stride (floats): 16B-aligned, bank-spread
#define X_PAD  132       // 32-row X slab row stride (floats): 16B-aligned, conflict-free

// ---------------------------------------------------------------------------
// Kernel 1: LDS-tiled WMMA GEMM.  Per batch: C(256x1024) = W(256x1024) *
// X(1024x1024pix) + bias.  Block = 64M x 128N, K-slab 32, 8 waves (2x4),
// each wave a 32x32 macro-tile = 2x2 v_wmma_f32_16x16x4_f32 accumulators.
// Coalesced b128 slab loads -> LDS; fragments via ds_load_b64/b32.
// Cuts L2 traffic ~5x vs. the untiled version (~192 MB total).
// ---------------------------------------------------------------------------
__global__ __launch_bounds__(256) void conv1x1_wmma(
    const float* __restrict__ feat,   // (8,1024,32,32)
    const float* __restrict__ Wt,     // (256,1024) row-major
    const float* __restrict__ bias,   // (256)
    float* __restrict__ xout)         // (8,256,1024)
{
    __shared__ float sW[BLK_M * W_PAD];   //  9.2 KB
    __shared__ float sX[BLK_K * X_PAD];   // 16.9 KB

    const int tid    = threadIdx.x;
    const int lane   = tid & 31;
    const int waveId = tid >> 5;          // 0..7
    const int wave_m = waveId >> 2;       // 0..1  (32-row strips)
    const int wave_n = waveId & 3;        // 0..3  (32-col strips)

    const int n_blk = blockIdx.x * BLK_N;     // 8 blocks over N=1024
    const int m_blk = blockIdx.y * BLK_M;     // 4 blocks over M=256
    const int b     = blockIdx.z;             // batch

    const float* __restrict__ Xb = feat + (size_t)b * (CIN * NPIX);

    // cooperative-load coordinates
    const int wrow = tid >> 3, wcol = (tid & 7) * 4;    // W: 32 rows/pass x 8 float4
    const int xrow = tid >> 5, xcol = (tid & 31) * 4;   // X:  8 rows/pass x 32 float4

    // fragment coordinates (ISA 32-bit A/B lane layouts)
    const int lane15 = lane & 15;
    const int koff   = (lane >> 4) << 1;                // 0 or 2

    v8f acc[2][2] = {};

    for (int k0 = 0; k0 < CIN; k0 += BLK_K) {
        __syncthreads();
        // W slab: 64 x 32 (2 passes of 32 rows)
        #pragma unroll
        for (int p = 0; p < 2; ++p) {
            const int r = wrow + p * 32;
            *(float4*)&sW[r * W_PAD + wcol] =
                *(const float4*)(Wt + (size_t)(m_blk + r) * CIN + k0 + wcol);
        }
        // X slab: 32 x 128 (4 passes of 8 rows)
        #pragma unroll
        for (int p = 0; p < 4; ++p) {
            const int r = xrow + p * 8;
            *(float4*)&sX[r * X_PAD + xcol] =
                *(const float4*)(Xb + (size_t)(k0 + r) * NPIX + n_blk + xcol);
        }
        __syncthreads();

        #pragma unroll
        for (int ks = 0; ks < BLK_K; ks += 4) {
            const int kl = ks + koff;
            // A fragments: rows (wave_m*32 + mt*16 + lane15), cols kl..kl+1
            v2f a0 = *(const v2f*)&sW[(wave_m * 32 +  0 + lane15) * W_PAD + kl];
            v2f a1 = *(const v2f*)&sW[(wave_m * 32 + 16 + lane15) * W_PAD + kl];
            // B fragments: rows kl..kl+1, cols (wave_n*32 + nt*16 + lane15)
            const int c0 = wave_n * 32 + lane15;
            v2f b0, b1;
            b0.x = sX[kl * X_PAD + c0];
            b0.y = sX[(kl + 1) * X_PAD + c0];
            b1.x = sX[kl * X_PAD + c0 + 16];
            b1.y = sX[(kl + 1) * X_PAD + c0 + 16];

            acc[0][0] = __builtin_amdgcn_wmma_f32_16x16x4_f32(
                false, a0, false, b0, (short)0, acc[0][0], false, false);
            acc[0][1] = __builtin_amdgcn_wmma_f32_16x16x4_f32(
                false, a0, false, b1, (short)0, acc[0][1], false, false);
            acc[1][0] = __builtin_amdgcn_wmma_f32_16x16x4_f32(
                false, a1, false, b0, (short)0, acc[1][0], false, false);
            acc[1][1] = __builtin_amdgcn_wmma_f32_16x16x4_f32(
                false, a1, false, b1, (short)0, acc[1][1], false, false);
        }
    }

    // C/D 16x16 f32 layout: lanes 0-15 N=lane, VGPR v = M=v ;
    //                       lanes 16-31 N=lane-16, VGPR v = M=8+v
    const int mhalf = (lane >> 4) << 3;
    #pragma unroll
    for (int mt = 0; mt < 2; ++mt) {
        #pragma unroll
        for (int nt = 0; nt < 2; ++nt) {
            const int m0 = m_blk + wave_m * 32 + mt * 16 + mhalf;
            const int n0 = n_blk + wave_n * 32 + nt * 16 + lane15;
            float* __restrict__ ocol = xout + (size_t)b * (COUT * NPIX) + n0;
            #pragma unroll
            for (int v = 0; v < 8; ++v) {
                const int m = m0 + v;
                ocol[(size_t)m * NPIX] = acc[mt][nt][v] + bias[m];
            }
        }
    }
}

// ---------------------------------------------------------------------------
// Kernel 2: mask downsample (exact point-sample: linspace step is exactly 33),
// masked pools -> feats(768), FC(6x768), softmax -> weights.  One block/batch.
// ---------------------------------------------------------------------------
__global__ __launch_bounds__(256) void pool_fc_softmax(
    const float* __restrict__ mask,   // (8,1,1024,1024)
    const float* __restrict__ xin,    // (8,256,1024) from kernel 1
    const float* __restrict__ fc_w,   // (6,768)
    const float* __restrict__ fc_b,   // (6)
    float* __restrict__ wout)         // (8,8) padded softmax weights
{
    __shared__ float fg[NPIX];
    __shared__ float feats[768];
    __shared__ float red[256];
    const int b = blockIdx.x, tid = threadIdx.x;

    for (int q = tid; q < NPIX; q += 256) {
        const int i = q >> 5, j = q & 31;
        fg[q] = mask[(size_t)b * HW_IMG + (size_t)(i * 33) * 1024 + j * 33];
    }
    __syncthreads();

    float s = 0.f;
    for (int q = tid; q < NPIX; q += 256) s += fg[q];
    red[tid] = s;
    __syncthreads();
    for (int off = 128; off > 0; off >>= 1) {
        if (tid < off) red[tid] += red[tid + off];
        __syncthreads();
    }
    const float Sfg      = red[0];
    const float inv_fg   = 1.0f / (Sfg + 1e-6f);
    const float inv_bg   = 1.0f / ((1024.0f - Sfg) + 1e-6f);   // sum(1-fg)
    const float inv_full = 1.0f / (1024.0f + 1e-6f);           // fg+bg == 1

    // channel tid: dot over 1024 pixels (float4, LDS-resident fg)
    const float4* __restrict__ xrow =
        (const float4*)(xin + (size_t)b * COUT * NPIX + (size_t)tid * NPIX);
    float afg = 0.f, afull = 0.f;
    for (int p4 = 0; p4 < NPIX / 4; ++p4) {
        const float4 v = xrow[p4];
        const int p = p4 * 4;
        afg   += v.x * fg[p] + v.y * fg[p + 1] + v.z * fg[p + 2] + v.w * fg[p + 3];
        afull += v.x + v.y + v.z + v.w;
    }
    feats[tid]       = afg * inv_fg;
    feats[256 + tid] = (afull - afg) * inv_bg;   // sum(x*(1-fg)) = afull - afg
    feats[512 + tid] = afull * inv_full;
    __syncthreads();

    if (tid < 6) {
        float lg = fc_b[tid];
        const float* __restrict__ wrow = fc_w + tid * 768;
        for (int k = 0; k < 768; ++k) lg += feats[k] * wrow[k];
        red[tid] = lg;
    }
    __syncthreads();
    if (tid == 0) {
        float mx = red[0];
        for (int l = 1; l < 6; ++l) mx = fmaxf(mx, red[l]);
        float e[6], se = 0.f;
        for (int l = 0; l < 6; ++l) { e[l] = expf(red[l] - mx); se += e[l]; }
        const float inv = 1.0f / se;
        for (int l = 0; l < 6; ++l) wout[b * 8 + l] = e[l] * inv;
    }
}

// ---------------------------------------------------------------------------
// Kernel 3: blended[b] = sum_l weights[b,l] * luts[l]  (3.45 MB, L2-resident)
// ---------------------------------------------------------------------------
__global__ __launch_bounds__(256) void blend_luts(
    const float* __restrict__ luts,     // (6,107811)
    const float* __restrict__ wts,      // (8,8)
    float* __restrict__ blended)        // (8,107811)
{
    const int idx = blockIdx.x * 256 + threadIdx.x;
    if (idx >= 8 * LUTC) return;
    const int b = idx / LUTC;
    const int r = idx - b * LUTC;
    const float* __restrict__ wb = wts + b * 8;
    float acc = 0.f;
    #pragma unroll
    for (int l = 0; l < 6; ++l) acc += wb[l] * luts[(size_t)l * LUTC + r];
    blended[idx] = acc;
}

// ---------------------------------------------------------------------------
// Kernel 4: trilinear LUT apply + mask composite.  4 pixels/thread with
// float4 (global_load/store_b128) on image/mask/out; 8-corner gathers hit the
// L2-resident blended LUT.  This is the bandwidth-bound kernel (~235 MB).
// ---------------------------------------------------------------------------
__global__ __launch_bounds__(256) void apply_lut(
    const float* __restrict__ image,    // (8,3,1024,1024)
    const float* __restrict__ mask,     // (8,1,1024,1024)
    const float* __restrict__ blended,  // (8,3,33,33,33)
    float* __restrict__ out)            // (8,3,1024,1024)
{
    const int gid = blockIdx.x * 256 + threadIdx.x;
    const int P  = gid << 2;            // first of 4 pixels
    const int b  = P >> 20;             // / (1024*1024)
    const int pp = P & (HW_IMG - 1);

    const float* __restrict__ ib = image + (size_t)b * 3 * HW_IMG + pp;
    const float4 r4 = *(const float4*)(ib);
    const float4 g4 = *(const float4*)(ib + HW_IMG);
    const float4 b4 = *(const float4*)(ib + 2 * HW_IMG);
    const float4 m4 = *(const float4*)(mask + (size_t)b * HW_IMG + pp);
    const float* __restrict__ lut = blended + (size_t)b * LUTC;

    const float INVBIN = 32.0f / 1.000001f;   // 1/BINSIZE
    float ro[4], go[4], bo[4];
    const float* rp = (const float*)&r4;
    const float* gp = (const float*)&g4;
    const float* bp = (const float*)&b4;
    const float* mp = (const float*)&m4;

    #pragma unroll
    for (int i = 0; i < 4; ++i) {
        const float r = rp[i], g = gp[i], bl = bp[i], m = mp[i];
        const float tr = r * INVBIN, tg = g * INVBIN, tb = bl * INVBIN;
        int ri = (int)tr; ri = ri < 0 ? 0 : (ri > 31 ? 31 : ri);
        int gi = (int)tg; gi = gi < 0 ? 0 : (gi > 31 ? 31 : gi);
        int bi = (int)tb; bi = bi < 0 ? 0 : (bi > 31 ? 31 : bi);
        const float fr = tr - (float)ri, fgw = tg - (float)gi, fb = tb - (float)bi;

        const int base = ri + gi * 33 + bi * 1089;
        const float w000 = (1.f - fr) * (1.f - fgw) * (1.f - fb);
        const float w100 = fr         * (1.f - fgw) * (1.f - fb);
        const float w010 = (1.f - fr) * fgw         * (1.f - fb);
        const float w110 = fr         * fgw         * (1.f - fb);
        const float w001 = (1.f - fr) * (1.f - fgw) * fb;
        const float w101 = fr         * (1.f - fgw) * fb;
        const float w011 = (1.f - fr) * fgw         * fb;
        const float w111 = fr         * fgw         * fb;

        float gen[3];
        #pragma unroll
        for (int c = 0; c < 3; ++c) {
            const float* __restrict__ L = lut + c * LUT3 + base;
            gen[c] = w000 * L[0]    + w100 * L[1]
                   + w010 * L[33]   + w110 * L[34]
                   + w001 * L[1089] + w101 * L[1090]
                   + w011 * L[1122] + w111 * L[1123];
        }
        const float im = 1.0f - m;
        ro[i] = gen[0] * m + r  * im;
        go[i] = gen[1] * m + g  * im;
        bo[i] = gen[2] * m + bl * im;
    }

    float* __restrict__ ob = out + (size_t)b * 3 * HW_IMG + pp;
    *(float4*)(ob)              = make_float4(ro[0], ro[1], ro[2], ro[3]);
    *(float4*)(ob + HW_IMG)     = make_float4(go[0], go[1], go[2], go[3]);
    *(float4*)(ob + 2 * HW_IMG) = make_float4(bo[0], bo[1], bo[2], bo[3]);
}

// ---------------------------------------------------------------------------
extern "C" void kernel_launch(void* const* d_in, const int* in_sizes, int n_in,
                              void* d_out, int out_size, void* d_ws, size_t ws_size,
                              hipStream_t stream) {
    const float* feat   = (const float*)d_in[0];   // encoder_feat
    const float* image  = (const float*)d_in[1];
    const float* mask   = (const float*)d_in[2];
    const float* conv_w = (const float*)d_in[3];
    const float* conv_b = (const float*)d_in[4];
    const float* fc_w   = (const float*)d_in[5];
    const float* fc_b   = (const float*)d_in[6];
    const float* luts   = (const float*)d_in[7];
    float* out = (float*)d_out;

    float* ws      = (float*)d_ws;
    float* x       = ws;                 // 8*256*1024   = 2,097,152 floats
    float* wts     = ws + 2097152;       // 8*8 padded softmax weights
    float* blended = ws + 2097216;       // 8*107811     =   862,488 floats

    // GEMM: grid = (N/128, M/64, batch)
    dim3 gemm_grid(NPIX / BLK_N, COUT / BLK_M, 8);
    conv1x1_wmma<<<gemm_grid, 256, 0, stream>>>(feat, conv_w, conv_b, x);
    pool_fc_softmax<<<8, 256, 0, stream>>>(mask, x, fc_w, fc_b, wts);
    blend_luts<<<(8 * LUTC + 255) / 256, 256, 0, stream>>>(luts, wts, blended);
    // 8M pixels / 4 per thread / 256 threads = 8192 blocks
    apply_lut<<<8192, 256, 0, stream>>>(image, mask, blended, out);
}